// GAT_35167192220484
// MI455X (gfx1250) — compile-verified
//
#include <hip/hip_runtime.h>
#include <hip/hip_bf16.h>

typedef __attribute__((ext_vector_type(2))) float v2f;
typedef __attribute__((ext_vector_type(8))) float v8f;

#define HID 96
#define NEG_SLOPE 0.2f

// ---------------- fp32 atomic max via int punning (works for all signs) ----
__device__ __forceinline__ void atomicMaxFloat(float* addr, float val) {
    if (val >= 0.0f)
        atomicMax((int*)addr, __float_as_int(val));
    else
        atomicMin((unsigned int*)addr, __float_as_uint(val));
}

// ---------------- h = X @ W via V_WMMA_F32_16X16X4_F32 ---------------------
// block = (32, 6): wave y computes output cols [16y, 16y+16).
// Each wave handles TWO 16-row tiles (M=32 per block) to reuse B fragments.
// A-row index is clamped (not masked): A row m only feeds D row m, and
// out-of-range D rows are simply not stored, so clamping is exact.
template <int K>
__global__ __launch_bounds__(192) void gemm_wmma_f32(
    const float* __restrict__ X, const float* __restrict__ W,
    float* __restrict__ H, int N) {
    const int lane = threadIdx.x;            // 0..31
    const int colb = threadIdx.y * 16;       // 0,16,...,80
    const int row0 = blockIdx.x * 32;
    const int mr   = lane & 15;              // A row / B,D col within tile
    const int hk   = (lane >> 4) * 2;        // K sub-offset per ISA layout
    const int n    = colb + mr;

    int r0 = row0 + mr;      if (r0 >= N) r0 = N - 1;
    int r1 = row0 + 16 + mr; if (r1 >= N) r1 = N - 1;
    const float* x0p = X + (size_t)r0 * K;
    const float* x1p = X + (size_t)r1 * K;

    v8f acc0 = {}, acc1 = {};
#pragma unroll
    for (int k = 0; k < K; k += 4) {
        v2f b;
        b.x = W[(k + hk)     * HID + n];
        b.y = W[(k + hk + 1) * HID + n];
        v2f a0, a1;
        a0.x = x0p[k + hk]; a0.y = x0p[k + hk + 1];
        a1.x = x1p[k + hk]; a1.y = x1p[k + hk + 1];
        acc0 = __builtin_amdgcn_wmma_f32_16x16x4_f32(
            false, a0, false, b, (short)0, acc0, false, false);
        acc1 = __builtin_amdgcn_wmma_f32_16x16x4_f32(
            false, a1, false, b, (short)0, acc1, false, false);
    }
    // D layout: VGPR v holds row v + 8*(lane>>4), col = colb + mr
    const int rb = (lane >> 4) << 3;
#pragma unroll
    for (int v = 0; v < 8; ++v) {
        int ra = row0 + rb + v;
        if (ra < N) H[(size_t)ra * HID + n] = acc0[v];
        int rbv = row0 + 16 + rb + v;
        if (rbv < N) H[(size_t)rbv * HID + n] = acc1[v];
    }
}

// ---------------- es[i] = h[i].as, ed[i] = h[i].ad (wave per node) ---------
__global__ void attn_logits(const float* __restrict__ H,
                            const float* __restrict__ as_,
                            const float* __restrict__ ad_,
                            float* __restrict__ es, float* __restrict__ ed, int N) {
    int node = (blockIdx.x * blockDim.x + threadIdx.x) >> 5;
    int lane = threadIdx.x & 31;
    if (node >= N) return;
    float s0 = 0.0f, s1 = 0.0f;
#pragma unroll
    for (int c = lane; c < HID; c += 32) {
        float h = H[(size_t)node * HID + c];
        s0 += h * as_[c];
        s1 += h * ad_[c];
    }
#pragma unroll
    for (int off = 16; off; off >>= 1) {
        s0 += __shfl_down(s0, off, 32);
        s1 += __shfl_down(s1, off, 32);
    }
    if (lane == 0) { es[node] = s0; ed[node] = s1; }
}

// ---------------- init m=-inf, s=0, agg=0 ----------------------------------
__global__ void init_layer(float* __restrict__ m, float* __restrict__ s,
                           float* __restrict__ agg, int N) {
    int idx = blockIdx.x * blockDim.x + threadIdx.x;
    if (idx < N) { m[idx] = -__builtin_inff(); s[idx] = 0.0f; }
    if (idx < N * HID) agg[idx] = 0.0f;
}

__global__ void init_final(float* __restrict__ m, float* __restrict__ s,
                           float* __restrict__ out, int N) {
    int idx = blockIdx.x * blockDim.x + threadIdx.x;
    if (idx < N) { m[idx] = -__builtin_inff(); s[idx] = 0.0f; out[idx] = 0.0f; }
}

// ---------------- pass A: e = lrelu(es[src]+ed[dst]); segment max ----------
__global__ void edge_max(const long long* __restrict__ ei,
                         const float* __restrict__ es, const float* __restrict__ ed,
                         float* __restrict__ ebuf, float* __restrict__ m,
                         int E, int N) {
    int idx = blockIdx.x * blockDim.x + threadIdx.x;
    if (idx >= E + N) return;
    long long s, d;
    if (idx < E) { s = ei[idx]; d = ei[E + idx]; } else { s = d = idx - E; }
    float e = es[s] + ed[d];
    e = e > 0.0f ? e : e * NEG_SLOPE;
    ebuf[idx] = e;
    atomicMaxFloat(m + d, e);
}

// ---------------- pass B: ex = exp(e - m[dst]); segment sum ----------------
__global__ void edge_exp(const long long* __restrict__ ei,
                         float* __restrict__ ebuf, const float* __restrict__ m,
                         float* __restrict__ sden, int E, int N) {
    int idx = blockIdx.x * blockDim.x + threadIdx.x;
    if (idx >= E + N) return;
    long long d = (idx < E) ? ei[E + idx] : (long long)(idx - E);
    float ex = __expf(ebuf[idx] - m[d]);
    ebuf[idx] = ex;
    unsafeAtomicAdd(sden + d, ex);        // native global_atomic_add_f32
}

// ---------------- pass C: agg[dst] += alpha * h[src] (wave per edge) -------
__global__ void edge_agg(const long long* __restrict__ ei,
                         const float* __restrict__ ebuf, const float* __restrict__ sden,
                         const float* __restrict__ H, float* __restrict__ agg,
                         int E, int N) {
    int gid  = blockIdx.x * (blockDim.x >> 5) + (threadIdx.x >> 5);
    int lane = threadIdx.x & 31;
    if (gid >= E + N) return;
    long long s, d;
    if (gid < E) { s = ei[gid]; d = ei[E + gid]; } else { s = d = gid - E; }
    float alpha = ebuf[gid] / sden[d];
    const float* hs = H + (size_t)s * HID;
    float* od = agg + (size_t)d * HID;
#pragma unroll
    for (int c = lane; c < HID; c += 32)
        unsafeAtomicAdd(od + c, alpha * hs[c]);
}

// ---------------- epilogue: out = relu(agg + b (+ residual)) ---------------
__global__ void epilogue(const float* __restrict__ agg, const float* __restrict__ b,
                         const float* __restrict__ res, float* __restrict__ xout,
                         int N) {
    int idx = blockIdx.x * blockDim.x + threadIdx.x;
    if (idx >= N * HID) return;
    int c = idx % HID;
    float v = agg[idx] + b[c];
    if (res) v += res[idx];
    xout[idx] = fmaxf(v, 0.0f);
}

// ---------------- final layer: hf = x.Wf ; es=hf*as ; ed=hf*ad -------------
__global__ void final_h(const float* __restrict__ X, const float* __restrict__ Wf,
                        const float* __restrict__ asf, const float* __restrict__ adf,
                        float* __restrict__ hf, float* __restrict__ es,
                        float* __restrict__ ed, int N) {
    int node = (blockIdx.x * blockDim.x + threadIdx.x) >> 5;
    int lane = threadIdx.x & 31;
    if (node >= N) return;
    float s0 = 0.0f;
#pragma unroll
    for (int c = lane; c < HID; c += 32)
        s0 += X[(size_t)node * HID + c] * Wf[c];
#pragma unroll
    for (int off = 16; off; off >>= 1) s0 += __shfl_down(s0, off, 32);
    if (lane == 0) {
        hf[node] = s0;
        es[node] = s0 * asf[0];
        ed[node] = s0 * adf[0];
    }
}

// ---------------- final aggregation (scalar per edge) ----------------------
__global__ void edge_agg1(const long long* __restrict__ ei,
                          const float* __restrict__ ebuf, const float* __restrict__ sden,
                          const float* __restrict__ hf, float* __restrict__ out,
                          int E, int N) {
    int idx = blockIdx.x * blockDim.x + threadIdx.x;
    if (idx >= E + N) return;
    long long s, d;
    if (idx < E) { s = ei[idx]; d = ei[E + idx]; } else { s = d = idx - E; }
    float alpha = ebuf[idx] / sden[d];
    unsafeAtomicAdd(out + d, alpha * hf[s]);
}

__global__ void final_bias(float* __restrict__ out, const float* __restrict__ bf, int N) {
    int idx = blockIdx.x * blockDim.x + threadIdx.x;
    if (idx < N) out[idx] += bf[0];
}

// ===========================================================================
extern "C" void kernel_launch(void* const* d_in, const int* in_sizes, int n_in,
                              void* d_out, int out_size, void* d_ws, size_t ws_size,
                              hipStream_t stream) {
    const int N = in_sizes[0] / 32;          // 50000
    const int E = in_sizes[1] / 2;           // 800000
    const int EN = E + N;

    const float*     x0 = (const float*)d_in[0];
    const long long* ei = (const long long*)d_in[1];
    // d_in[2] = edge_weight (ignored by reference)
    const float* Ws[3] = {(const float*)d_in[3], (const float*)d_in[7],  (const float*)d_in[11]};
    const float* As[3] = {(const float*)d_in[4], (const float*)d_in[8],  (const float*)d_in[12]};
    const float* Ad[3] = {(const float*)d_in[5], (const float*)d_in[9],  (const float*)d_in[13]};
    const float* Bs[3] = {(const float*)d_in[6], (const float*)d_in[10], (const float*)d_in[14]};
    const float* Wf  = (const float*)d_in[15];
    const float* asf = (const float*)d_in[16];
    const float* adf = (const float*)d_in[17];
    const float* bf  = (const float*)d_in[18];
    float* out = (float*)d_out;

    // workspace carve-up (floats)
    float* ws = (float*)d_ws;
    size_t off = 0;
    float* h    = ws + off; off += (size_t)N * HID;
    float* xA   = ws + off; off += (size_t)N * HID;
    float* xB   = ws + off; off += (size_t)N * HID;
    float* agg  = ws + off; off += (size_t)N * HID;
    float* es   = ws + off; off += (size_t)N;
    float* ed   = ws + off; off += (size_t)N;
    float* m    = ws + off; off += (size_t)N;
    float* sden = ws + off; off += (size_t)N;
    float* ebuf = ws + off; off += (size_t)EN;

    const int TPB = 256;
    dim3 gemmBlk(32, 6, 1);
    int gemmGrid   = (N + 31) / 32;          // 32 rows per block
    int nodeWaves  = (N + 7) / 8;            // wave-per-node kernels
    int nhGrid     = (N * HID + TPB - 1) / TPB;
    int nGrid      = (N + TPB - 1) / TPB;
    int eGrid      = (EN + TPB - 1) / TPB;
    int eWaveGrid  = (EN + 7) / 8;           // wave-per-edge kernel

    const float* x_in = x0;
    float* xbufs[2] = {xA, xB};

    for (int l = 0; l < 3; ++l) {
        float* x_out = xbufs[l & 1];
        if (l == 0)
            gemm_wmma_f32<32><<<gemmGrid, gemmBlk, 0, stream>>>(x_in, Ws[l], h, N);
        else
            gemm_wmma_f32<96><<<gemmGrid, gemmBlk, 0, stream>>>(x_in, Ws[l], h, N);
        attn_logits<<<nodeWaves, TPB, 0, stream>>>(h, As[l], Ad[l], es, ed, N);
        init_layer<<<nhGrid, TPB, 0, stream>>>(m, sden, agg, N);
        edge_max<<<eGrid, TPB, 0, stream>>>(ei, es, ed, ebuf, m, E, N);
        edge_exp<<<eGrid, TPB, 0, stream>>>(ei, ebuf, m, sden, E, N);
        edge_agg<<<eWaveGrid, TPB, 0, stream>>>(ei, ebuf, sden, h, agg, E, N);
        // residual only when dims match (layers 1,2); ReLU always
        epilogue<<<nhGrid, TPB, 0, stream>>>(agg, Bs[l], l == 0 ? nullptr : x_in, x_out, N);
        x_in = x_out;
    }

    // final layer (96 -> 1): hf reuses h buffer (needs only N floats)
    final_h<<<nodeWaves, TPB, 0, stream>>>(x_in, Wf, asf, adf, h, es, ed, N);
    init_final<<<nGrid, TPB, 0, stream>>>(m, sden, out, N);
    edge_max<<<eGrid, TPB, 0, stream>>>(ei, es, ed, ebuf, m, E, N);
    edge_exp<<<eGrid, TPB, 0, stream>>>(ei, ebuf, m, sden, E, N);
    edge_agg1<<<eGrid, TPB, 0, stream>>>(ei, ebuf, sden, h, out, E, N);
    final_bias<<<nGrid, TPB, 0, stream>>>(out, bf, N);
}